// SessionGNN_38087769981120
// MI455X (gfx1250) — compile-verified
//
#include <hip/hip_runtime.h>

typedef __attribute__((ext_vector_type(2))) float v2f;
typedef __attribute__((ext_vector_type(8))) float v8f;

#define NUM_USERS 50000
#define NUM_JOBS  50000
#define NUM_NODES 100000
#define HID 64
#define SCAN_BLK 1024

// D = A(16x4 f32) * B(4x16 f32) + C(16x16 f32), wave32 WMMA.
__device__ __forceinline__ v8f wmma4_f32(v2f a, v2f b, v8f c) {
    return __builtin_amdgcn_wmma_f32_16x16x4_f32(
        /*neg_a=*/false, a, /*neg_b=*/false, b,
        /*c_mod=*/(short)0, c, /*reuse_a=*/false, /*reuse_b=*/false);
}

// Pair-interleave a K x 64 weight matrix: element (k,col) ->
// Wp[((k/2)*64 + col)*2 + (k&1)], so {W[k][col], W[k+1][col]} (k even) is one
// aligned 8-byte pair -> single b64 load straight into a WMMA B-fragment.
__global__ void repack_pairs_kernel(const float* __restrict__ W,
                                    float* __restrict__ Wp, int n /* K*64 */) {
    int i = blockIdx.x * blockDim.x + threadIdx.x;
    if (i >= n) return;
    const int k = i >> 6;
    const int col = i & 63;
    Wp[((size_t)((k >> 1) << 6) + col) * 2 + (k & 1)] = W[i];
}

// Fused two-layer dense: out = (act(feat @ W1 + b1)) @ W2 + b2
// feat: ntiles*16 x K1 row-major. W1p: pair-interleaved K1 x 64.
// W2: 64 x 64 row-major (interleaved into LDS here). out: ntiles*16 x 64.
template<int K1, bool RELU1>
__launch_bounds__(256)
__global__ void fused2_kernel(const float* __restrict__ feat,
                              const float* __restrict__ W1p,
                              const float* __restrict__ b1,
                              const float* __restrict__ W2,
                              const float* __restrict__ b2,
                              float* __restrict__ out,
                              int ntiles) {
    __shared__ float ldsW2[64 * 64];       // 16 KB, pair-interleaved
    __shared__ float ldsC1[8 * 16 * 64];   // 32 KB: per-wave 16x64 staging

    for (int i = threadIdx.x; i < 64 * 64; i += 256) {
        const int k = i >> 6, col = i & 63;
        ldsW2[(((k >> 1) << 6) + col) * 2 + (k & 1)] = W2[i];
    }
    __syncthreads();

    const int wave = threadIdx.x >> 5;
    const int lane = threadIdx.x & 31;
    const int grp  = lane >> 4;   // 0|1: selects K pair (A/B) and M half (C/D)
    const int nn   = lane & 15;   // A: row m; B/C: col n
    const int tile = blockIdx.x * 8 + wave;
    if (tile >= ntiles) return;   // wave-uniform: EXEC stays all-1s for WMMA

    const int m0 = tile * 16;
    float* c1 = ldsC1 + wave * (16 * 64);

    // ---- GEMM 1: 16 x K1 x 64 ----
    v8f acc[4];
    #pragma unroll
    for (int t = 0; t < 4; ++t)
        #pragma unroll
        for (int j = 0; j < 8; ++j) acc[t][j] = 0.0f;

    for (int k0 = 0; k0 < K1; k0 += 4) {
        const int ka = k0 + 2 * grp;             // even
        const v2f a = *(const v2f*)(feat + (size_t)(m0 + nn) * K1 + ka);
        #pragma unroll
        for (int t = 0; t < 4; ++t) {
            const v2f b = *(const v2f*)(
                W1p + ((size_t)((ka >> 1) << 6) + t * 16 + nn) * 2);
            acc[t] = wmma4_f32(a, b, acc[t]);
        }
    }

    // bias (+relu), stage C1 tile into per-wave LDS in row-major form
    #pragma unroll
    for (int t = 0; t < 4; ++t) {
        const int col = t * 16 + nn;
        const float bb = b1[col];
        #pragma unroll
        for (int j = 0; j < 8; ++j) {
            const int m = j + 8 * grp;
            float v = acc[t][j] + bb;
            if (RELU1) v = fmaxf(v, 0.0f);
            c1[m * 64 + col] = v;
        }
    }

    // ---- GEMM 2: 16 x 64 x 64 ----
    v8f acc2[4];
    #pragma unroll
    for (int t = 0; t < 4; ++t)
        #pragma unroll
        for (int j = 0; j < 8; ++j) acc2[t][j] = 0.0f;

    #pragma unroll
    for (int k0 = 0; k0 < 64; k0 += 4) {
        const int ka = k0 + 2 * grp;
        const v2f a = *(const v2f*)(c1 + nn * 64 + ka);   // adjacent -> ds_load_b64
        #pragma unroll
        for (int t = 0; t < 4; ++t) {
            const v2f b = *(const v2f*)(
                ldsW2 + ((((ka >> 1) << 6) + t * 16 + nn) * 2));
            acc2[t] = wmma4_f32(a, b, acc2[t]);
        }
    }

    #pragma unroll
    for (int t = 0; t < 4; ++t) {
        const int col = t * 16 + nn;
        const float bb = b2[col];
        #pragma unroll
        for (int j = 0; j < 8; ++j) {
            const int m = j + 8 * grp;
            out[(size_t)(m0 + m) * HID + col] = acc2[t][j] + bb;
        }
    }
}

// xw = x @ W  (ntiles*16 x 64 x 64), no bias. W row-major (interleaved here).
__launch_bounds__(256)
__global__ void gemm_xw_kernel(const float* __restrict__ x,
                               const float* __restrict__ W,
                               float* __restrict__ xw,
                               int ntiles) {
    __shared__ float ldsW[64 * 64];   // pair-interleaved
    for (int i = threadIdx.x; i < 64 * 64; i += 256) {
        const int k = i >> 6, col = i & 63;
        ldsW[(((k >> 1) << 6) + col) * 2 + (k & 1)] = W[i];
    }
    __syncthreads();

    const int wave = threadIdx.x >> 5;
    const int lane = threadIdx.x & 31;
    const int grp  = lane >> 4;
    const int nn   = lane & 15;
    const int tile = blockIdx.x * 8 + wave;
    if (tile >= ntiles) return;

    const int m0 = tile * 16;
    v8f acc[4];
    #pragma unroll
    for (int t = 0; t < 4; ++t)
        #pragma unroll
        for (int j = 0; j < 8; ++j) acc[t][j] = 0.0f;

    #pragma unroll
    for (int k0 = 0; k0 < 64; k0 += 4) {
        const int ka = k0 + 2 * grp;
        const v2f a = *(const v2f*)(x + (size_t)(m0 + nn) * 64 + ka);
        #pragma unroll
        for (int t = 0; t < 4; ++t) {
            const v2f b = *(const v2f*)(
                ldsW + ((((ka >> 1) << 6) + t * 16 + nn) * 2));
            acc[t] = wmma4_f32(a, b, acc[t]);
        }
    }

    #pragma unroll
    for (int t = 0; t < 4; ++t) {
        const int col = t * 16 + nn;
        #pragma unroll
        for (int j = 0; j < 8; ++j) {
            const int m = j + 8 * grp;
            xw[(size_t)(m0 + m) * 64 + col] = acc[t][j];
        }
    }
}

// -------------------- CSR construction --------------------

__global__ void zero_int_kernel(int* __restrict__ p, int n) {
    int i = blockIdx.x * blockDim.x + threadIdx.x;
    if (i < n) p[i] = 0;
}

__global__ void count_kernel(const int* __restrict__ dst,
                             int* __restrict__ cnt, int nE) {
    int i = blockIdx.x * blockDim.x + threadIdx.x;
    if (i < nE) atomicAdd(&cnt[dst[i]], 1);
}

// dinv[i] = rsqrt(in_degree + 1)  (self-loop included)
__global__ void dinv_kernel(const int* __restrict__ cnt,
                            float* __restrict__ dinv, int n) {
    int i = blockIdx.x * blockDim.x + threadIdx.x;
    if (i < n) dinv[i] = rsqrtf((float)cnt[i] + 1.0f);
}

// Per-block exclusive scan (Hillis-Steele in LDS); block totals to bsum.
__launch_bounds__(SCAN_BLK)
__global__ void scan_block_kernel(const int* __restrict__ cnt,
                                  int* __restrict__ rowstart,
                                  int* __restrict__ bsum, int n) {
    __shared__ int tmp[SCAN_BLK];
    const int tid = threadIdx.x;
    const int gid = blockIdx.x * SCAN_BLK + tid;
    const int v = (gid < n) ? cnt[gid] : 0;
    tmp[tid] = v;
    __syncthreads();
    #pragma unroll
    for (int off = 1; off < SCAN_BLK; off <<= 1) {
        int t = (tid >= off) ? tmp[tid - off] : 0;
        __syncthreads();
        tmp[tid] += t;
        __syncthreads();
    }
    if (gid < n) rowstart[gid] = tmp[tid] - v;   // exclusive
    if (tid == SCAN_BLK - 1) bsum[blockIdx.x] = tmp[tid];
}

__global__ void scan_tops_kernel(int* __restrict__ bsum, int nb) {
    if (blockIdx.x == 0 && threadIdx.x == 0) {
        int acc = 0;
        for (int i = 0; i < nb; ++i) { int v = bsum[i]; bsum[i] = acc; acc += v; }
    }
}

__global__ void finalize_rows_kernel(int* __restrict__ rowstart,
                                     int* __restrict__ cursor,
                                     const int* __restrict__ bsum,
                                     int n, int nE) {
    int i = blockIdx.x * blockDim.x + threadIdx.x;
    if (i < n) {
        int v = rowstart[i] + bsum[i >> 10];
        rowstart[i] = v;
        cursor[i] = v;
    }
    if (i == 0) rowstart[n] = nE;
}

__global__ void fill_kernel(const int* __restrict__ src,
                            const int* __restrict__ dst,
                            int* __restrict__ cursor,
                            int* __restrict__ csr_src, int nE) {
    int i = blockIdx.x * blockDim.x + threadIdx.x;
    if (i < nE) {
        int p = atomicAdd(&cursor[dst[i]], 1);
        csr_src[p] = src[i];
    }
}

// -------------------- GCN gather (atomic-free) --------------------
// One wave per destination node; 2 channels per lane (64 total).
// xout[d] = relu( dinv[d]^2*xw[d] + sum_e dinv[src]*dinv[d]*xw[src] + b )
__launch_bounds__(256)
__global__ void gcn_gather_kernel(const float* __restrict__ xw,
                                  const float* __restrict__ dinv,
                                  const int* __restrict__ rowstart,
                                  const int* __restrict__ csr_src,
                                  const float* __restrict__ b,
                                  float* __restrict__ xout, int nnodes) {
    const int node = (int)((blockIdx.x * blockDim.x + threadIdx.x) >> 5);
    const int lane = threadIdx.x & 31;
    if (node >= nnodes) return;
    const int c = lane * 2;
    const float dd = dinv[node];

    const float2 self = *(const float2*)(xw + (size_t)node * 64 + c);
    float a0 = dd * dd * self.x;
    float a1 = dd * dd * self.y;

    const int e0 = rowstart[node];
    const int e1 = rowstart[node + 1];
    for (int e = e0; e < e1; ++e) {
        const int s = csr_src[e];                // wave-uniform broadcast load
        const float w = dd * dinv[s];
        const float2 v = *(const float2*)(xw + (size_t)s * 64 + c);
        a0 += w * v.x;
        a1 += w * v.y;
    }
    float* o = xout + (size_t)node * 64 + c;
    o[0] = fmaxf(a0 + b[c + 0], 0.0f);
    o[1] = fmaxf(a1 + b[c + 1], 0.0f);
}

extern "C" void kernel_launch(void* const* d_in, const int* in_sizes, int n_in,
                              void* d_out, int out_size, void* d_ws, size_t ws_size,
                              hipStream_t stream) {
    const float* user_features = (const float*)d_in[0];
    const float* job_features  = (const float*)d_in[1];
    const int*   edge_index    = (const int*)d_in[2];
    const float* uw1 = (const float*)d_in[3];
    const float* ub1 = (const float*)d_in[4];
    const float* uw2 = (const float*)d_in[5];
    const float* ub2 = (const float*)d_in[6];
    const float* jw1 = (const float*)d_in[7];
    const float* jb1 = (const float*)d_in[8];
    const float* jw2 = (const float*)d_in[9];
    const float* jb2 = (const float*)d_in[10];
    const float* gw1 = (const float*)d_in[11];
    const float* gb1 = (const float*)d_in[12];
    const float* gw2 = (const float*)d_in[13];
    const float* gb2 = (const float*)d_in[14];
    // d_in[15..18] = wq,bq,wk,bk : unused by the reference output
    const float* wv = (const float*)d_in[19];
    const float* bv = (const float*)d_in[20];
    const float* wo = (const float*)d_in[21];
    const float* bo = (const float*)d_in[22];
    float* out = (float*)d_out;

    const int nE = in_sizes[2] / 2;
    const int* src = edge_index;
    const int* dst = edge_index + nE;

    // ---- workspace layout ----
    float* x    = (float*)d_ws;                       // NUM_NODES*64
    float* xw   = x + (size_t)NUM_NODES * 64;         // NUM_NODES*64
    float* dinv = xw + (size_t)NUM_NODES * 64;        // NUM_NODES
    float* uw1p = dinv + NUM_NODES;                   // 256*64 pair-interleaved
    float* jw1p = uw1p + 256 * 64;                    // 256*64
    float* wvp  = jw1p + 256 * 64;                    // 64*64
    int* ip       = (int*)(wvp + 64 * 64);
    int* cnt      = ip;                               // NUM_NODES
    int* rowstart = cnt + NUM_NODES;                  // NUM_NODES+1
    int* cursor   = rowstart + NUM_NODES + 1;         // NUM_NODES
    int* csr_src  = cursor + NUM_NODES;               // nE
    int* bsum     = csr_src + nE;                     // scan tops (<=128)

    const int tilesU = NUM_USERS / 16;   // 3125
    const int tilesN = NUM_NODES / 16;   // 6250
    const int nscan  = (NUM_NODES + SCAN_BLK - 1) / SCAN_BLK;  // 98
    dim3 blk(256);

    // Weight repack (one-shot, tiny)
    repack_pairs_kernel<<<(256 * 64 + 255) / 256, blk, 0, stream>>>(uw1, uw1p, 256 * 64);
    repack_pairs_kernel<<<(256 * 64 + 255) / 256, blk, 0, stream>>>(jw1, jw1p, 256 * 64);
    repack_pairs_kernel<<<(64 * 64 + 255) / 256, blk, 0, stream>>>(wv, wvp, 64 * 64);

    // Encoders (WMMA)
    fused2_kernel<256, true><<<(tilesU + 7) / 8, blk, 0, stream>>>(
        user_features, uw1p, ub1, uw2, ub2, x, tilesU);
    fused2_kernel<256, true><<<(tilesU + 7) / 8, blk, 0, stream>>>(
        job_features, jw1p, jb1, jw2, jb2, x + (size_t)NUM_USERS * 64, tilesU);

    // CSR build (int atomics only, done once)
    zero_int_kernel<<<(NUM_NODES + 255) / 256, blk, 0, stream>>>(cnt, NUM_NODES);
    count_kernel<<<(nE + 255) / 256, blk, 0, stream>>>(dst, cnt, nE);
    dinv_kernel<<<(NUM_NODES + 255) / 256, blk, 0, stream>>>(cnt, dinv, NUM_NODES);
    scan_block_kernel<<<nscan, SCAN_BLK, 0, stream>>>(cnt, rowstart, bsum, NUM_NODES);
    scan_tops_kernel<<<1, 32, 0, stream>>>(bsum, nscan);
    finalize_rows_kernel<<<(NUM_NODES + 255) / 256, blk, 0, stream>>>(
        rowstart, cursor, bsum, NUM_NODES, nE);
    fill_kernel<<<(nE + 255) / 256, blk, 0, stream>>>(src, dst, cursor, csr_src, nE);

    // Two GCN layers: WMMA GEMM + atomic-free register-accumulate gather
    for (int layer = 0; layer < 2; ++layer) {
        const float* W = layer ? gw2 : gw1;
        const float* b = layer ? gb2 : gb1;
        gemm_xw_kernel<<<(tilesN + 7) / 8, blk, 0, stream>>>(x, W, xw, tilesN);
        gcn_gather_kernel<<<(NUM_NODES * 32 + 255) / 256, blk, 0, stream>>>(
            xw, dinv, rowstart, csr_src, b, x, NUM_NODES);
    }

    // Head: ((u @ wv + bv) @ wo + bo)  -- no relu between (WMMA)
    fused2_kernel<64, false><<<(tilesU + 7) / 8, blk, 0, stream>>>(
        x, wvp, bv, wo, bo, out, tilesU);
}